// QRDcell_34557306864068
// MI455X (gfx1250) — compile-verified
//
#include <hip/hip_runtime.h>
#include <hip/hip_bf16.h>
#include <math.h>

// ---------------------------------------------------------------------------
// QR-iteration scan: per batch b, serial over t:
//   t < 5 :  out[b,t,:] = 0, Q unchanged (Q starts as I)
//   t >= 5:  M = J[b,t] @ Q ; [Qn,R] = qr(M) ; out[b,t,:] = diag(R); Q = Qn
// One workgroup (256 threads = 8 wave32 waves) per batch element.
//  - M = Jt @ Q via V_WMMA_F32_16X16X4_F32 (only fp32 WMMA on CDNA5).
//  - QR via deferred-normalization modified Gram-Schmidt in LDS,
//    4-way row-group parallel dot products (16-deep loops).
//  - Next Jt staged with GLOBAL_LOAD_ASYNC_TO_LDS_B128 (ASYNCcnt),
//    overlapped with the whole MGS phase; sync fallback if builtin absent.
// ---------------------------------------------------------------------------

#define NU      64
#define LDSS    65          // LDS row stride (odd -> conflict-free column walks)
#define START_T 5

typedef float v2f __attribute__((ext_vector_type(2)));
typedef float v8f __attribute__((ext_vector_type(8)));
typedef int   ivec4 __attribute__((vector_size(16)));

#if defined(__HIP_DEVICE_COMPILE__) && \
    __has_builtin(__builtin_amdgcn_global_load_async_to_lds_b128)
#define HAVE_ASYNC_LDS 1
#else
#define HAVE_ASYNC_LDS 0
#endif

__device__ __forceinline__ void issue_tile_load(const float* __restrict__ Jt,
                                                float* __restrict__ Js, int tid) {
#if HAVE_ASYNC_LDS
    // 64x64 f32 tile = 1024 x 16B chunks; 4 async b128 per thread.
    #pragma unroll
    for (int it = 0; it < 4; ++it) {
        const int chunk = tid + it * 256;
        const int r = chunk >> 4;           // row (16 chunks per row)
        const int c = (chunk & 15) << 2;    // column of first float
        __builtin_amdgcn_global_load_async_to_lds_b128(
            (__attribute__((address_space(1))) ivec4*)(Jt + r * NU + c),
            (__attribute__((address_space(3))) ivec4*)(&Js[r * LDSS + c]),
            0, 0);
    }
#else
    (void)Jt; (void)Js; (void)tid;
#endif
}

__device__ __forceinline__ void stage_sync(const float* __restrict__ Jt,
                                           float* __restrict__ Js, int tid) {
    const float4* Jv = (const float4*)Jt;
    for (int i = tid; i < NU * NU / 4; i += 256) {
        float4 v = Jv[i];
        const int r = (i * 4) >> 6, c = (i * 4) & 63;
        float* p = &Js[r * LDSS + c];
        p[0] = v.x; p[1] = v.y; p[2] = v.z; p[3] = v.w;
    }
}

__global__ __launch_bounds__(256, 1)
void qrd_scan_kernel(const float* __restrict__ J, float* __restrict__ out, int T) {
    __shared__ float Qs[NU * LDSS];   // carried orthonormal Q
    __shared__ float Ms[NU * LDSS];   // M = Jt @ Q, MGS-transformed in place
    __shared__ float Js[NU * LDSS];   // staged Jt tile
    __shared__ float part[4 * NU];    // row-group partial dot products
    __shared__ float dvec[NU];        // d_k = m_j . m_{j+kk}
    __shared__ float invn[NU];        // deferred column normalizers d_j^{-1/2}
    __shared__ float Rdiag[NU];       // diag(R)
    __shared__ float scal[1];         // 1/d_j

    const int b    = blockIdx.x;
    const int tid  = threadIdx.x;
    const int lane = tid & 31;
    const int wave = tid >> 5;
    const int half = lane >> 4;       // 0: lanes 0-15, 1: lanes 16-31
    const int mrow = lane & 15;

    // Q = I
    for (int i = tid; i < NU * NU; i += 256) {
        const int r = i >> 6, c = i & 63;
        Qs[r * LDSS + c] = (r == c) ? 1.0f : 0.0f;
    }
    __syncthreads();

    const float* Jb = J + (size_t)b * (size_t)T * NU * NU;

#if HAVE_ASYNC_LDS
    if (T > START_T)                              // pre-issue first tile
        issue_tile_load(Jb + (size_t)START_T * NU * NU, Js, tid);
#endif

    for (int t = 0; t < T; ++t) {
        if (t < START_T) {                        // uniform across the block
            if (tid < NU) out[((size_t)b * T + t) * NU + tid] = 0.0f;
            continue;
        }
        const float* Jt = Jb + (size_t)t * NU * NU;

#if HAVE_ASYNC_LDS
        asm volatile("s_wait_asynccnt 0x0" ::: "memory");   // my waves' tile landed
        __syncthreads();                                    // everyone's tile landed
#else
        if (t + 1 < T) __builtin_prefetch(Jt + NU * NU + tid * 16, 0, 0);
        stage_sync(Jt, Js, tid);
        __syncthreads();
#endif

        // ------- M = Jt @ Q via v_wmma_f32_16x16x4_f32 -------
        // 4x4 grid of 16x16 output tiles; wave w owns tiles w and w+8.
        for (int rep = 0; rep < 2; ++rep) {
            const int tile = wave + rep * 8;
            const int ti = tile >> 2, tj = tile & 3;
            v8f acc = {};
            for (int kk = 0; kk < 16; ++kk) {
                const int k0 = kk * 4 + 2 * half;  // A: lanes 0-15 K=0,1; 16-31 K=2,3
                v2f a, bf;
                a[0]  = Js[(ti * 16 + mrow) * LDSS + k0 + 0];
                a[1]  = Js[(ti * 16 + mrow) * LDSS + k0 + 1];
                bf[0] = Qs[(k0 + 0) * LDSS + tj * 16 + mrow];
                bf[1] = Qs[(k0 + 1) * LDSS + tj * 16 + mrow];
                acc = __builtin_amdgcn_wmma_f32_16x16x4_f32(
                          false, a, false, bf, (short)0, acc, false, false);
            }
            // C/D layout: VGPR r -> row r (lanes 0-15) / row r+8 (lanes 16-31)
            for (int r = 0; r < 8; ++r)
                Ms[(ti * 16 + r + 8 * half) * LDSS + tj * 16 + mrow] = acc[r];
        }
        __syncthreads();                          // all waves done reading Js

#if HAVE_ASYNC_LDS
        if (t + 1 < T)                            // overlap next tile with MGS
            issue_tile_load(Jt + NU * NU, Js, tid);
#endif

        // ------- QR via deferred-normalization MGS -------
        // Per j: d_k = m_j.m_k (k>=j) in one fused pass (4 row-groups x 64 cols),
        // then m_k -= (d_k/d_j) m_j. Normalization deferred to the Q-carry copy.
        for (int j = 0; j < NU; ++j) {
            {   // Phase A: partial dots
                const int kk = tid & 63;          // column offset: k = j + kk
                const int rg = tid >> 6;          // row group 0..3
                const int k  = j + kk;
                float s = 0.0f;
                if (k < NU) {
                    const int r0 = rg * 16;
                    #pragma unroll
                    for (int r = 0; r < 16; ++r)
                        s += Ms[(r0 + r) * LDSS + j] * Ms[(r0 + r) * LDSS + k];
                }
                part[rg * NU + kk] = s;
            }
            __syncthreads();

            if (tid < NU) {                       // Phase A2: 4-way reduce
                const float d = part[tid] + part[NU + tid] +
                                part[2 * NU + tid] + part[3 * NU + tid];
                dvec[tid] = d;
                if (tid == 0) {
                    Rdiag[j] = sqrtf(d);
                    invn[j]  = (d > 0.0f) ? rsqrtf(d)   : 0.0f;
                    scal[0]  = (d > 0.0f) ? (1.0f / d)  : 0.0f;
                }
            }
            __syncthreads();

            {   // Phase B: rank-1 update with unnormalized m_j
                const float idj = scal[0];
                for (int i = tid; i < NU * NU; i += 256) {
                    const int r = i >> 6, c = i & 63;
                    if (c > j)
                        Ms[r * LDSS + c] -= (dvec[c - j] * idj) * Ms[r * LDSS + j];
                }
            }
            __syncthreads();
        }

        // Emit diag(R); carry Q <- M * diag(invn) (deferred normalization).
        if (tid < NU) out[((size_t)b * T + t) * NU + tid] = Rdiag[tid];
        for (int i = tid; i < NU * NU; i += 256) {
            const int r = i >> 6, c = i & 63;
            Qs[r * LDSS + c] = Ms[r * LDSS + c] * invn[c];
        }
        __syncthreads();                          // Qs ready for next step's WMMA
    }
}

extern "C" void kernel_launch(void* const* d_in, const int* in_sizes, int n_in,
                              void* d_out, int out_size, void* d_ws, size_t ws_size,
                              hipStream_t stream) {
    (void)n_in; (void)d_ws; (void)ws_size; (void)out_size;
    const float* J = (const float*)d_in[0];
    float* out     = (float*)d_out;

    // in_sizes[0] = B*T*64*64
    const int BT = in_sizes[0] / (NU * NU);
    const int T  = 200;                 // per reference
    const int B  = BT / T;

    qrd_scan_kernel<<<dim3(B), dim3(256), 0, stream>>>(J, out, T);
}